// CELossWeighted_28698971472547
// MI455X (gfx1250) — compile-verified
//
#include <hip/hip_runtime.h>
#include <hip/hip_bf16.h>

// ---------------------------------------------------------------------------
// Weighted cross-entropy loss, MI455X (gfx1250).
// Memory-bound: ~144MB traffic -> ~6.2us roofline @ 23.3 TB/s.
// Main pass streams x through LDS with CDNA5 async copies
// (global_load_async_to_lds_b128, double buffered, s_wait_asynccnt).
// Final reduction uses V_WMMA_F32_16X16X4_F32 (ones-matrix row-sum trick).
// All reductions deterministic (integer atomics only; float partials summed
// in fixed order).
// ---------------------------------------------------------------------------

typedef __attribute__((ext_vector_type(2))) float v2f;
typedef __attribute__((ext_vector_type(8))) float v8f;

#define NCH        32
#define HW         65536        // 256*256
#define NPIX       1048576      // 16*256*256
#define TILE       128          // pixels per tile
#define NTILES     (NPIX / TILE)   // 8192
#define MAIN_BLOCKS 512
#define MAIN_THREADS 128

// d_ws layout (bytes):
//   [0,128)        : counts, 32 x u32
//   [512, 2560)    : partial numerators,  512 x f32
//   [2560, 4608)   : partial denominators,512 x f32

__global__ void __launch_bounds__(32)
zero_counts_kernel(unsigned* __restrict__ counts) {
    counts[threadIdx.x & 31] = 0u;
}

__global__ void __launch_bounds__(256)
hist_kernel(const int* __restrict__ tgt, unsigned* __restrict__ counts) {
    __shared__ unsigned h[NCH];
    const int tid = threadIdx.x;
    if (tid < NCH) h[tid] = 0u;
    __syncthreads();
    const int stride = gridDim.x * blockDim.x;
    for (int i = blockIdx.x * blockDim.x + tid; i < NPIX; i += stride)
        atomicAdd(&h[tgt[i] & 31], 1u);
    __syncthreads();
    if (tid < NCH) atomicAdd(&counts[tid], h[tid]);
}

__global__ void __launch_bounds__(MAIN_THREADS)
ce_main_kernel(const float* __restrict__ x, const int* __restrict__ tgt,
               const unsigned* __restrict__ counts,
               float* __restrict__ pnum, float* __restrict__ pden) {
    // 2 buffers x 32 channels x 128 pixels x 4B = 32KB
    __shared__ __align__(16) float s_x[2 * NCH * TILE];
    __shared__ float s_recip[NCH];
    __shared__ float s_rn[4], s_rd[4];

    const int tid = threadIdx.x;
    if (tid < NCH) s_recip[tid] = 1.0f / (float)counts[tid];

    // Per-thread copy descriptors: 8 x b128 chunks per tile.
    // chunk idx = tid + j*128 in [0,1024); c = idx/32, k = idx%32.
    // Generic LDS pointer: low 32 bits are the raw LDS byte address (ISA 10.2).
    const unsigned ldsbase = (unsigned)(size_t)(&s_x[0]);
    int      ckoff[8];
    unsigned ldsoff[8];
#pragma unroll
    for (int j = 0; j < 8; ++j) {
        const int idx = tid + j * TILE;
        const int c = idx >> 5;
        const int k = idx & 31;
        ckoff[j]  = c * (HW * 4) + k * 16;                       // global byte off
        ldsoff[j] = ldsbase + (unsigned)(c * (TILE * 4) + k * 16); // lds byte addr
    }
    const unsigned long long gx = (unsigned long long)x;

    float lnum = 0.0f, lden = 0.0f;

    int t = blockIdx.x;
    { // prologue: prefetch first tile into buffer 0
        const int p0 = t << 7;
        const int tb = ((p0 >> 16) << 23) + ((p0 & (HW - 1)) << 2);
#pragma unroll
        for (int j = 0; j < 8; ++j)
            asm volatile("global_load_async_to_lds_b128 %0, %1, %2"
                         :: "v"(ldsoff[j]), "v"(tb + ckoff[j]), "s"(gx)
                         : "memory");
    }

    unsigned buf = 0;
    for (; t < NTILES; t += MAIN_BLOCKS) {
        const int tn = t + MAIN_BLOCKS;
        if (tn < NTILES) {
            // prefetch next tile into the other buffer
            const int p0 = tn << 7;
            const int tb = ((p0 >> 16) << 23) + ((p0 & (HW - 1)) << 2);
            const unsigned bsel = (buf ^ 1u) * (NCH * TILE * 4);
#pragma unroll
            for (int j = 0; j < 8; ++j)
                asm volatile("global_load_async_to_lds_b128 %0, %1, %2"
                             :: "v"(ldsoff[j] + bsel), "v"(tb + ckoff[j]), "s"(gx)
                             : "memory");
            // ASYNCcnt completes in order -> outstanding ops are a suffix of
            // issue order; <=8 outstanding means the previous tile landed.
            asm volatile("s_wait_asynccnt 8" ::: "memory");
        } else {
            asm volatile("s_wait_asynccnt 0" ::: "memory");
        }
        __syncthreads();   // all waves' async data visible; recip table ready

        { // compute: one pixel per thread; LDS [c][pixel] is conflict-free
            const float* col = &s_x[buf * (NCH * TILE) + tid];
            const int targ = tgt[t * TILE + tid] & 31;
            float m = col[0];
#pragma unroll
            for (int c = 1; c < NCH; ++c) m = fmaxf(m, col[c * TILE]);
            float ssum = 0.0f;
#pragma unroll
            for (int c = 0; c < NCH; ++c) ssum += __expf(col[c * TILE] - m);
            const float xt = col[targ * TILE];           // dynamic LDS gather
            const float ce = m + __logf(ssum) - xt;      // -log_softmax[targ]
            const float w  = s_recip[targ];
            lnum = __builtin_fmaf(ce, w, lnum);
            lden += w;
        }
        __syncthreads();   // everyone done reading buf before it is refilled
        buf ^= 1u;
    }

    // block reduction (wave32 shuffle tree, then 4 waves via LDS)
#pragma unroll
    for (int off = 16; off > 0; off >>= 1) {
        lnum += __shfl_xor(lnum, off, 32);
        lden += __shfl_xor(lden, off, 32);
    }
    const int wid = tid >> 5, lane = tid & 31;
    if (lane == 0) { s_rn[wid] = lnum; s_rd[wid] = lden; }
    __syncthreads();
    if (tid == 0) {
        pnum[blockIdx.x] = s_rn[0] + s_rn[1] + s_rn[2] + s_rn[3];
        pden[blockIdx.x] = s_rd[0] + s_rd[1] + s_rd[2] + s_rd[3];
    }
}

// Final reduction: one wave. 512 partials per stream, folded through
// V_WMMA_F32_16X16X4_F32 with an all-ones B: D[m][n] += sum_k A[m][k].
// A layout (16x4 f32): lanes 0-15 hold (M=lane, K=0,1); lanes 16-31 (M=lane-16, K=2,3).
__global__ void __launch_bounds__(32)
final_wmma_kernel(const float* __restrict__ pnum, const float* __restrict__ pden,
                  float* __restrict__ out) {
    const int lane = threadIdx.x;
    v8f cn = {0.f,0.f,0.f,0.f,0.f,0.f,0.f,0.f};
    v8f cd = {0.f,0.f,0.f,0.f,0.f,0.f,0.f,0.f};
    v2f ones; ones.x = 1.0f; ones.y = 1.0f;
    const int base0 = ((lane & 15) << 2) + ((lane >> 4) << 1);
#pragma unroll
    for (int i = 0; i < 8; ++i) {
        const int b = i * 64 + base0;
        v2f an; an.x = pnum[b]; an.y = pnum[b + 1];
        v2f ad; ad.x = pden[b]; ad.y = pden[b + 1];
        cn = __builtin_amdgcn_wmma_f32_16x16x4_f32(false, an, false, ones,
                                                   (short)0, cn, false, false);
        cd = __builtin_amdgcn_wmma_f32_16x16x4_f32(false, ad, false, ones,
                                                   (short)0, cd, false, false);
    }
    // C/D layout: lane l holds rows {0..7}+8*(l>=16) at column l%16.
    float sn = 0.0f, sd = 0.0f;
#pragma unroll
    for (int k = 0; k < 8; ++k) { sn += cn[k]; sd += cd[k]; }
    sn += __shfl_xor(sn, 16, 32);  // rows 0-7 + rows 8-15 = column total
    sd += __shfl_xor(sd, 16, 32);
    if (lane == 0) out[0] = sn / sd;
}

extern "C" void kernel_launch(void* const* d_in, const int* in_sizes, int n_in,
                              void* d_out, int out_size, void* d_ws, size_t ws_size,
                              hipStream_t stream) {
    const float* x   = (const float*)d_in[0];
    const int*   tgt = (const int*)d_in[1];

    unsigned* counts = (unsigned*)d_ws;
    float*    wsf    = (float*)d_ws;
    float*    pnum   = wsf + 128;        // byte offset 512
    float*    pden   = wsf + 128 + 512;  // byte offset 2560
    float*    out    = (float*)d_out;

    zero_counts_kernel<<<1, 32, 0, stream>>>(counts);
    hist_kernel<<<256, 256, 0, stream>>>(tgt, counts);
    ce_main_kernel<<<MAIN_BLOCKS, MAIN_THREADS, 0, stream>>>(x, tgt, counts, pnum, pden);
    final_wmma_kernel<<<1, 32, 0, stream>>>(pnum, pden, out);
}